// BasicLinear_module_73607149518999
// MI455X (gfx1250) — compile-verified
//
#include <hip/hip_runtime.h>

// CDNA5 / gfx1250 fused Linear(64->64) + bias + row-LayerNorm + LeakyReLU(0.2)
// fp32 end-to-end using V_WMMA_F32_16X16X4_F32. Memory-bound (~1 GB traffic,
// ~46us floor at 23.3 TB/s); fp32 WMMA throughput is not the bottleneck.

typedef float v2f __attribute__((ext_vector_type(2)));
typedef float v4f __attribute__((ext_vector_type(4)));
typedef float v8f __attribute__((ext_vector_type(8)));
typedef int   v4i __attribute__((ext_vector_type(4)));

typedef __attribute__((address_space(1))) v4i* gptr_v4i;   // global
typedef __attribute__((address_space(3))) v4i* lptr_v4i;   // LDS

#define IN_F 64
#define OUT_F 64
#define ROWS_PER_BLOCK 128
#define THREADS 256
// 68 dwords = 272 B row stride: 16B-aligned rows (b128 LDS ops) and bank-
// conflict-free for both A and B fragment b64 reads (banks 4r+{0..3}).
#define XSTRIDE 68

#if defined(__has_builtin)
#if __has_builtin(__builtin_amdgcn_global_load_async_to_lds_b128)
#define HAVE_ASYNC_LDS 1
#endif
#if __has_builtin(__builtin_amdgcn_s_wait_asynccnt)
#define HAVE_WAIT_ASYNC 1
#endif
#endif

__global__ __launch_bounds__(THREADS)
void fused_linear_ln_lrelu(const float* __restrict__ x,
                           const float* __restrict__ W,
                           const float* __restrict__ bias,
                           const float* __restrict__ gamma,
                           const float* __restrict__ beta,
                           float* __restrict__ out,
                           int n_nodes, int n_tiles)
{
    __shared__ float sX[ROWS_PER_BLOCK * XSTRIDE];   // 34816 B
    __shared__ float sW[OUT_F * XSTRIDE];            // 17408 B

    const int tid  = threadIdx.x;
    const int wave = tid >> 5;
    const int lane = tid & 31;
    const int c    = lane & 15;   // col-within-16 (B/C/D) and row-within-16 (A)
    const int h    = lane >> 4;   // half-wave select (K pair / M+8)

    // ---- stage W once per block: sW[o][k] = W[o][k] (torch layout W[out,in]) ----
    for (int u = tid; u < (OUT_F * IN_F) / 4; u += THREADS) {
        v4f w4 = ((const v4f*)W)[u];
        *(v4f*)&sW[(u >> 4) * XSTRIDE + ((u & 15) << 2)] = w4;   // ds_store_b128
    }

    // per-lane epilogue constants for the 4 output col tiles
    float g[4], bt[4], bs[4];
#pragma unroll
    for (int t = 0; t < 4; ++t) {
        int col = t * 16 + c;
        g[t]  = gamma[col];
        bt[t] = beta[col];
        bs[t] = bias[col];
    }

    for (int tile = blockIdx.x; tile < n_tiles; tile += gridDim.x) {
        const long rowBase  = (long)tile * ROWS_PER_BLOCK;
        const bool fullTile = (rowBase + ROWS_PER_BLOCK) <= (long)n_nodes;

        __syncthreads();   // protect sX from previous iteration's readers

        // ---- stage 128x64 x-tile into LDS ----
        if (fullTile) {
#ifdef HAVE_ASYNC_LDS
            // CDNA5 async copy: global -> LDS directly, no VGPR round-trip.
            for (int u = tid; u < (ROWS_PER_BLOCK * IN_F) / 4; u += THREADS) {
                int row = u >> 4;
                int kq  = (u & 15) << 2;
                const float* src = x + (rowBase + row) * IN_F + kq;
                float*       dst = &sX[row * XSTRIDE + kq];
                __builtin_amdgcn_global_load_async_to_lds_b128(
                    (gptr_v4i)(void*)const_cast<float*>(src),
                    (lptr_v4i)(void*)dst,
                    /*offset=*/0, /*cpol=*/0);
            }
#ifdef HAVE_WAIT_ASYNC
            __builtin_amdgcn_s_wait_asynccnt(0);
#else
            asm volatile("s_wait_asynccnt 0" ::: "memory");
#endif
#else
            for (int u = tid; u < (ROWS_PER_BLOCK * IN_F) / 4; u += THREADS) {
                int row = u >> 4;
                int kq  = (u & 15) << 2;
                v4f v = __builtin_nontemporal_load(((const v4f*)x) +
                                                   (rowBase + row) * 16 + (u & 15));
                *(v4f*)&sX[row * XSTRIDE + kq] = v;   // ds_store_b128
            }
#endif
        } else {
            for (int u = tid; u < (ROWS_PER_BLOCK * IN_F) / 4; u += THREADS) {
                int row = u >> 4;
                int kq  = (u & 15) << 2;
                long grow = rowBase + row;
                v4f v = {0.f, 0.f, 0.f, 0.f};
                if (grow < (long)n_nodes)
                    v = __builtin_nontemporal_load(((const v4f*)x) + grow * 16 + (u & 15));
                *(v4f*)&sX[row * XSTRIDE + kq] = v;
            }
        }
        __syncthreads();

        // ---- GEMM: wave handles node rows [16*wave, 16*wave+16) ----
        const int r0 = wave * 16;
        v8f acc[4];
#pragma unroll
        for (int t = 0; t < 4; ++t) {
            v8f a;
#pragma unroll
            for (int i = 0; i < 8; ++i) a[i] = bs[t];  // fold bias into C
            acc[t] = a;
        }

        const float* aRow = &sX[(r0 + c) * XSTRIDE];
#pragma unroll
        for (int kb = 0; kb < IN_F; kb += 4) {
            // A fragment: x[r0+c][kb + 2h + {0,1}]
            v2f afrag = *(const v2f*)&aRow[kb + 2 * h];
#pragma unroll
            for (int t = 0; t < 4; ++t) {
                // B fragment: B[k][n] = W[n][k], n = 16t + c, k = kb + 2h + {0,1}
                v2f bfrag = *(const v2f*)&sW[(t * 16 + c) * XSTRIDE + kb + 2 * h];
                acc[t] = __builtin_amdgcn_wmma_f32_16x16x4_f32(
                    /*neg_a=*/false, afrag, /*neg_b=*/false, bfrag,
                    /*c_mod=*/(short)0, acc[t],
                    /*reuse_a=*/false, /*reuse_b=*/false);
            }
        }

        // ---- fused LayerNorm + LeakyReLU epilogue, straight from registers ----
        // acc[t][v] = y[row r0 + v + 8h][col 16t + c]; a row's 64 values live in
        // one VGPR index across 4 tiles x one 16-lane half of the wave.
#pragma unroll
        for (int v = 0; v < 8; ++v) {
            float s = 0.f, q = 0.f;
#pragma unroll
            for (int t = 0; t < 4; ++t) {
                float y = acc[t][v];
                s += y;
                q += y * y;
            }
#pragma unroll
            for (int m = 1; m < 16; m <<= 1) {   // reduce within 16-lane half
                s += __shfl_xor(s, m, 32);
                q += __shfl_xor(q, m, 32);
            }
            float mu   = s * (1.0f / 64.0f);
            float var  = q * (1.0f / 64.0f) - mu * mu;
            float rinv = rsqrtf(var + 1e-5f);

            long grow = rowBase + r0 + v + 8 * h;
            float* orow = out + grow * 64;
            if (fullTile) {
#pragma unroll
                for (int t = 0; t < 4; ++t) {
                    float y = (acc[t][v] - mu) * rinv * g[t] + bt[t];
                    y = (y >= 0.f) ? y : 0.2f * y;
                    __builtin_nontemporal_store(y, orow + t * 16 + c);
                }
            } else if (grow < (long)n_nodes) {
#pragma unroll
                for (int t = 0; t < 4; ++t) {
                    float y = (acc[t][v] - mu) * rinv * g[t] + bt[t];
                    y = (y >= 0.f) ? y : 0.2f * y;
                    __builtin_nontemporal_store(y, orow + t * 16 + c);
                }
            }
        }
    }
}

extern "C" void kernel_launch(void* const* d_in, const int* in_sizes, int n_in,
                              void* d_out, int out_size, void* d_ws, size_t ws_size,
                              hipStream_t stream)
{
    // setup_inputs order: input_x, W, b, gamma, beta, batch (batch unused: the
    // per-graph layernorm is mathematically row-wise, independent of batch ids)
    const float* x     = (const float*)d_in[0];
    const float* W     = (const float*)d_in[1];
    const float* b     = (const float*)d_in[2];
    const float* gamma = (const float*)d_in[3];
    const float* beta  = (const float*)d_in[4];
    float* out         = (float*)d_out;

    int n_nodes = in_sizes[0] / IN_F;
    int n_tiles = (n_nodes + ROWS_PER_BLOCK - 1) / ROWS_PER_BLOCK;
    int grid    = n_tiles < 4096 ? n_tiles : 4096;   // persistent blocks, grid-stride

    fused_linear_ln_lrelu<<<grid, THREADS, 0, stream>>>(x, W, b, gamma, beta, out,
                                                        n_nodes, n_tiles);
}